// GlassboxDiffKAN_11218454577802
// MI455X (gfx1250) — compile-verified
//
#include <hip/hip_runtime.h>
#include <hip/hip_bf16.h>

// CDNA5 / gfx1250 fused KAN kernel, v2.
//
// out[b] = w.h + h^T S h   with  h = P c(b),  g = (S P) c(b) + w
//        = sum_v h_v * g_v
//
// WMMA mapping (V_WMMA_F32_16X16X4_F32, K = 48 in 12 chunks):
//   A (16 x 48): rows 0-7 = P, rows 8-15 = S*P. Stationary (24 VGPRs).
//   B (48 x 16): candidates, N = batch column. B layout wants lo lanes to
//     supply K = 4kk+{0,1} and hi lanes K = 4kk+{2,3} of column n = lane&15.
//     Each lane therefore loads BOTH rows of its xor-16 pair (r0 = wbase+n,
//     r1 = r0+16; same cachelines wave-wide) and computes exactly the 24
//     candidate values its lane-half feeds -- no cross-lane traffic in the
//     hot loop.
//   C init: 0 in the h-half (lanes 0-15), w[v] in the g-half (lanes 16-31).
// After the chains, lane n holds h_*[n], lane n+16 holds g_*[n]:
//   out[n] = sum_v acc_v * swizzle_xor16(acc_v)   (ds_swizzle SWAPX16).
// Four independent 6-deep WMMA chains per wave (2 tiles x split-K).
//
// K index map (k = i*6+op): k mod 4 in {0,1} (lo) / {2,3} (hi) resolves to:
//   per input pair (xe = x[2g], xo = x[2g+1]), with A-set = {x,sin,x^2,sigm},
//   B-set = {cos,tanh}:
//     lo: slots 3g..3g+2 = (xe:x,sin), (xe:x^2,sigm), (xo:cos,tanh)
//     hi: slots 3g..3g+2 = (xe:cos,tanh), (xo:x,sin), (xo:x^2,sigm)
//   i.e. A-set input = lo?xe:xo, B-set input = lo?xo:xe, slots rotated by 1.

typedef float v2f __attribute__((ext_vector_type(2)));
typedef float v8f __attribute__((ext_vector_type(8)));

__device__ __forceinline__ float fast_rcp(float x) {
    return __builtin_amdgcn_rcpf(x);
}

__device__ __forceinline__ float swz16(float x) {  // lane L <-> lane L^16
    return __int_as_float(__builtin_amdgcn_ds_swizzle(__float_as_int(x), 0x401F));
}

// Build the 12 v2f B-operand slots this lane contributes for one row.
__device__ __forceinline__ void cand_slots(const float4 x0, const float4 x1,
                                           bool lo, v2f d[12]) {
    const float xin[8] = {x0.x, x0.y, x0.z, x0.w, x1.x, x1.y, x1.z, x1.w};
    #pragma unroll
    for (int g = 0; g < 4; ++g) {
        const float xe = xin[2 * g], xo = xin[2 * g + 1];
        const float xa = lo ? xe : xo;           // feeds {x, sin, x^2, sigmoid}
        const float xb = lo ? xo : xe;           // feeds {cos, tanh}
        const float sa = __sinf(xa);
        const float sg = fast_rcp(1.f + __expf(-xa));        // sigmoid
        const float qa = xa * xa;
        const float cb = __cosf(xb);
        const float u  = __expf(-2.f * xb);
        const float tb = (1.f - u) * fast_rcp(1.f + u);      // tanh
        // P = (xa, sin), Q = (xa^2, sigm), R = (cos, tanh); hi rotates slots.
        d[3 * g + 0][0] = lo ? xa : cb;
        d[3 * g + 0][1] = lo ? sa : tb;
        d[3 * g + 1][0] = lo ? qa : xa;
        d[3 * g + 1][1] = lo ? sg : sa;
        d[3 * g + 2][0] = lo ? cb : qa;
        d[3 * g + 2][1] = lo ? tb : sg;
    }
}

// ---------------------------------------------------------------------------
// Setup kernel: build AW[16][48] = [ P ; S*P ] into workspace (single block).
// ---------------------------------------------------------------------------
__global__ __launch_bounds__(256) void kan_setup_weights(
    const float* __restrict__ ew,   // [8][8][6] edge_weights
    const float* __restrict__ gw,   // [P] gate_w
    const float* __restrict__ sc1,  // [P] scale1
    const float* __restrict__ sc2,  // [P] scale2
    const int*   __restrict__ pi,   // [P] pair_i
    const int*   __restrict__ pj,   // [P] pair_j
    int nPairs,
    float* __restrict__ AW)         // [16][48] output
{
    __shared__ float Pl[8 * 48];
    __shared__ float S[64];
    const int t = threadIdx.x;

    if (t < 64) {                                 // softmax over K=6 per (h,i)
        const int h = t >> 3, i = t & 7;
        const float* e = ew + h * 48 + i * 6;
        float mx = e[0];
        #pragma unroll
        for (int k = 1; k < 6; ++k) mx = fmaxf(mx, e[k]);
        float ex[6]; float s = 0.f;
        #pragma unroll
        for (int k = 0; k < 6; ++k) { ex[k] = __expf(e[k] - mx); s += ex[k]; }
        const float inv = fast_rcp(s);
        #pragma unroll
        for (int k = 0; k < 6; ++k) {
            const float p = ex[k] * inv;
            Pl[h * 48 + i * 6 + k] = p;
            AW[h * 48 + i * 6 + k] = p;
        }
        S[t] = 0.f;
    }
    __syncthreads();

    if (t == 0) {                                 // symmetric pair matrix S
        for (int p = 0; p < nPairs; ++p) {
            const int i = pi[p], j = pj[p];
            const float sig = fast_rcp(1.f + __expf(-gw[p]));
            const float cf = sig * sc1[p] * sc2[p];
            if (i == j) {
                S[i * 8 + i] += cf;
            } else {
                S[i * 8 + j] += 0.5f * cf;
                S[j * 8 + i] += 0.5f * cf;
            }
        }
    }
    __syncthreads();

    for (int idx = t; idx < 8 * 48; idx += blockDim.x) {   // rows 8-15 = S*P
        const int m = idx / 48, k = idx - m * 48;
        float acc = 0.f;
        #pragma unroll
        for (int j = 0; j < 8; ++j) acc += S[m * 8 + j] * Pl[j * 48 + k];
        AW[(8 + m) * 48 + k] = acc;
    }
}

// ---------------------------------------------------------------------------
// Main fused kernel.
// ---------------------------------------------------------------------------
__global__ __launch_bounds__(256) void kan_fused_wmma(
    const float* __restrict__ x,    // [B][8]
    const float* __restrict__ lw,   // [8] linear_w
    const float* __restrict__ AW,   // [16][48] from setup
    float* __restrict__ out,        // [B]
    int B)
{
    const int lane = threadIdx.x & 31;
    const bool lo  = lane < 16;
    const long long wbase = (long long)blockIdx.x * blockDim.x + (threadIdx.x & ~31);
    const long long e  = wbase + lane;                       // element stored
    long long r0 = wbase + (lane & 15);                      // tile0 column row
    long long r1 = r0 + 16;                                  // tile1 column row
    r0 = (r0 < B) ? r0 : (long long)B - 1;
    r1 = (r1 < B) ? r1 : (long long)B - 1;

    // Both rows of this lane's xor-16 pair (same cachelines wave-wide).
    const float4* xr0 = (const float4*)(x + r0 * 8);
    const float4* xr1 = (const float4*)(x + r1 * 8);
    const float4 a00 = xr0[0], a01 = xr0[1];
    const float4 b00 = xr1[0], b01 = xr1[1];

    // This lane's B-operand slots for the two tiles (no shuffles needed).
    v2f dA[12], dB[12];
    cand_slots(a00, a01, lo, dA);
    cand_slots(b00, b01, lo, dB);

    // Stationary A operand: lane L<16 -> (M=L, K=4kk+{0,1}); hi -> K=4kk+{2,3}.
    const int m  = lane & 15;
    const int hk = (lane >> 4) << 1;
    v2f a[12];
    #pragma unroll
    for (int kk = 0; kk < 12; ++kk)
        a[kk] = *(const v2f*)(AW + m * 48 + kk * 4 + hk);

    // Split-K accumulators: bias-carrying half + zero half, per tile.
    v8f acc0a, acc0b, acc1a, acc1b;
    #pragma unroll
    for (int v = 0; v < 8; ++v) {
        const float wv = lo ? 0.f : lw[v];   // g-half gets +w bias
        acc0a[v] = wv; acc1a[v] = wv;
        acc0b[v] = 0.f; acc1b[v] = 0.f;
    }

    // Four independent 6-deep WMMA chains, round-robin.
    #pragma unroll
    for (int kk = 0; kk < 6; ++kk) {
        acc0a = __builtin_amdgcn_wmma_f32_16x16x4_f32(
            false, a[kk], false, dA[kk], (short)0, acc0a, false, false);
        acc1a = __builtin_amdgcn_wmma_f32_16x16x4_f32(
            false, a[kk], false, dB[kk], (short)0, acc1a, false, false);
        acc0b = __builtin_amdgcn_wmma_f32_16x16x4_f32(
            false, a[kk + 6], false, dA[kk + 6], (short)0, acc0b, false, false);
        acc1b = __builtin_amdgcn_wmma_f32_16x16x4_f32(
            false, a[kk + 6], false, dB[kk + 6], (short)0, acc1b, false, false);
    }

    // out[n] = sum_v h_v[n] * g_v[n]; h and g sit 16 lanes apart.
    float s0 = 0.f, s1 = 0.f;
    #pragma unroll
    for (int v = 0; v < 8; ++v) {
        const float c0 = acc0a[v] + acc0b[v];
        const float c1 = acc1a[v] + acc1b[v];
        s0 += c0 * swz16(c0);
        s1 += c1 * swz16(c1);
    }
    const float res = lo ? s0 : s1;   // lane t owns element wbase + t

    if (e < B) out[e] = res;
}

// ---------------------------------------------------------------------------
// Host launch
// ---------------------------------------------------------------------------
extern "C" void kernel_launch(void* const* d_in, const int* in_sizes, int n_in,
                              void* d_out, int out_size, void* d_ws, size_t ws_size,
                              hipStream_t stream) {
    const float* x   = (const float*)d_in[0];   // [B,8]
    const float* ew  = (const float*)d_in[1];   // [8,8,6]
    const float* lw  = (const float*)d_in[2];   // [1,8]
    const float* gw  = (const float*)d_in[3];   // [36]
    const float* sc1 = (const float*)d_in[4];   // [36]
    const float* sc2 = (const float*)d_in[5];   // [36]
    const int*   pi  = (const int*)d_in[6];     // [36]
    const int*   pj  = (const int*)d_in[7];     // [36]
    float* out = (float*)d_out;

    const int B      = in_sizes[0] / 8;
    const int nPairs = in_sizes[3];
    float* AW = (float*)d_ws;                   // 16*48 floats = 3 KB

    kan_setup_weights<<<1, 256, 0, stream>>>(ew, gw, sc1, sc2, pi, pj, nPairs, AW);

    const int blocks = (B + 255) / 256;
    kan_fused_wmma<<<blocks, 256, 0, stream>>>(x, lw, AW, out, B);
}